// GATv2EncoderGated_25331717111857
// MI455X (gfx1250) — compile-verified
//
#include <hip/hip_runtime.h>
#include <cmath>

// ---------------------------------------------------------------------------
// GATv2 gated encoder for MI455X (gfx1250, wave32).
//   - Dense node GEMMs run on V_WMMA_F32_16X16X4_F32 (exact fp32 matrix pipe),
//     with batched fragment loads (8 K-steps per burst) so one s_wait covers
//     8 back-to-back WMMAs.
//   - Edge phase is L2-resident (xl+xr = 102MB < 192MB L2); edge projection
//     (edge_attr @ We) is recomputed on the fly instead of storing 819MB.
//   - Segment softmax via native float atomics (global_atomic_max_num_f32 /
//     global_atomic_add_f32) through unsafeAtomic*.
// ---------------------------------------------------------------------------

typedef __attribute__((ext_vector_type(2))) float v2f;
typedef __attribute__((ext_vector_type(8))) float v8f;

#define HEADS 4
#define HID   64
#define HC    256   // HEADS * HID

__device__ __forceinline__ float lrelu(float x, float s) { return x > 0.f ? x : s * x; }

// ---------------------------------------------------------------------------
// out[N x 256] = act(A[N x K]) @ W[K x 256] + bias
// One wave computes one 16x16 C tile; K stepped by 4 via WMMA f32 16x16x4.
// A frag (16x4 f32): lane = 16*k_hi + m, vgpr j holds K = 2*k_hi + j.
// B frag (4x16 f32): lane = 16*k_hi + n, vgpr j holds K = 2*k_hi + j.
// C frag: vgpr v, lane l -> M = v + 8*(l>>4), N = l & 15.
// A rows are clamped (not exec-masked) for the tail tile; stores are guarded.
// ---------------------------------------------------------------------------
template <int KDIM, bool LEAKY>
__global__ __launch_bounds__(256) void gemm_wmma_kernel(
    const float* __restrict__ A, const float* __restrict__ W,
    const float* __restrict__ bias, float* __restrict__ out, int Nn)
{
  const int lane    = threadIdx.x & 31;
  const int wave    = threadIdx.x >> 5;
  const int rowBase = blockIdx.x * 16;
  const int cb      = blockIdx.y * 128 + wave * 16;   // 8 waves cover 128 cols
  const int n       = lane & 15;
  const int kp      = (lane >> 4) << 1;               // 0 or 2

  int arow = rowBase + n;                             // A frag: M = lane & 15
  if (arow >= Nn) arow = Nn - 1;                      // clamp: no exec masking

  const float* __restrict__ Ap = A + (size_t)arow * KDIM + kp;
  const float* __restrict__ Bp = W + (size_t)kp * HC + cb + n;

  v8f acc = {};
  #pragma unroll
  for (int chunk = 0; chunk < KDIM / 32; ++chunk) {
    v2f a[8], b[8];
    // Burst-load 8 K-steps of A and B fragments (one clause, one wait).
    #pragma unroll
    for (int j = 0; j < 8; ++j) {
      const int k0 = chunk * 32 + j * 4;
      a[j].x = Ap[k0];
      a[j].y = Ap[k0 + 1];
      b[j].x = Bp[(size_t)k0 * HC];
      b[j].y = Bp[(size_t)(k0 + 1) * HC];
    }
    if (LEAKY) {
      #pragma unroll
      for (int j = 0; j < 8; ++j) {
        a[j].x = lrelu(a[j].x, 0.01f);
        a[j].y = lrelu(a[j].y, 0.01f);
      }
    }
    // 8 back-to-back WMMAs on the loaded fragments.
    #pragma unroll
    for (int j = 0; j < 8; ++j) {
      acc = __builtin_amdgcn_wmma_f32_16x16x4_f32(
          /*neg_a=*/false, a[j], /*neg_b=*/false, b[j],
          /*c_mod=*/(short)0, acc, /*reuse_a=*/false, /*reuse_b=*/false);
    }
  }

  const float bv    = bias[cb + n];
  const int   mbase = (lane >> 4) << 3;
  #pragma unroll
  for (int v = 0; v < 8; ++v) {
    const int r = rowBase + mbase + v;
    if (r < Nn) out[r * HC + cb + n] = acc[v] + bv;
  }
}

// accum[N*256]=0, mx[N*4]=-inf, denom[N*4]=0
__global__ __launch_bounds__(256) void init_kernel(
    float* __restrict__ accum, float* __restrict__ mx,
    float* __restrict__ denom, int Nn)
{
  const int i = blockIdx.x * blockDim.x + threadIdx.x;
  if (i < Nn * HC) accum[i] = 0.f;
  if (i < Nn * HEADS) { mx[i] = -INFINITY; denom[i] = 0.f; }
}

// One thread per (edge, head): logit = att_h . leaky(xl[src]+xr[dst]+eproj, 0.2)
// eproj recomputed from edge_attr (9) x We column slice (L0-resident, 9KB).
__global__ __launch_bounds__(256) void edge_logits_kernel(
    const float* __restrict__ xl, const float* __restrict__ xr,
    const float* __restrict__ eattr, const float* __restrict__ We,
    const float* __restrict__ att, const long long* __restrict__ ei,
    int Ecnt, float* __restrict__ logits, float* __restrict__ mx)
{
  const int t = blockIdx.x * blockDim.x + threadIdx.x;
  if (t >= Ecnt * HEADS) return;
  const int h = t & 3;
  const int e = t >> 2;
  const int src = (int)ei[e];
  const int dst = (int)ei[Ecnt + e];

  float ea[9];
  #pragma unroll
  for (int k = 0; k < 9; ++k) ea[k] = eattr[e * 9 + k];

  const float4* pl = reinterpret_cast<const float4*>(xl + src * HC + h * HID);
  const float4* pr = reinterpret_cast<const float4*>(xr + dst * HC + h * HID);
  const float4* pa = reinterpret_cast<const float4*>(att + h * HID);
  const float*  wb = We + h * HID;

  float acc = 0.f;
  #pragma unroll 4
  for (int c4 = 0; c4 < HID / 4; ++c4) {
    const float4 vl = pl[c4], vr = pr[c4], va = pa[c4];
    float ex = 0.f, ey = 0.f, ez = 0.f, ew = 0.f;
    #pragma unroll
    for (int k = 0; k < 9; ++k) {
      const float4 w = *reinterpret_cast<const float4*>(wb + k * HC + c4 * 4);
      ex = fmaf(ea[k], w.x, ex); ey = fmaf(ea[k], w.y, ey);
      ez = fmaf(ea[k], w.z, ez); ew = fmaf(ea[k], w.w, ew);
    }
    acc = fmaf(va.x, lrelu(vl.x + vr.x + ex, 0.2f), acc);
    acc = fmaf(va.y, lrelu(vl.y + vr.y + ey, 0.2f), acc);
    acc = fmaf(va.z, lrelu(vl.z + vr.z + ez, 0.2f), acc);
    acc = fmaf(va.w, lrelu(vl.w + vr.w + ew, 0.2f), acc);
  }
  logits[t] = acc;
  unsafeAtomicMax(&mx[dst * HEADS + h], acc);   // global_atomic_max_num_f32
}

// mx = where(isfinite(mx), mx, 0)
__global__ __launch_bounds__(256) void fix_mx_kernel(float* __restrict__ mx, int Nn)
{
  const int i = blockIdx.x * blockDim.x + threadIdx.x;
  if (i >= Nn * HEADS) return;
  const float v = mx[i];
  if (!__builtin_isfinite(v)) mx[i] = 0.f;
}

// p = exp(logit - mx[dst]); denom[dst] += p  (p overwrites logits buffer)
__global__ __launch_bounds__(256) void edge_exp_kernel(
    float* __restrict__ logits, const float* __restrict__ mx,
    float* __restrict__ denom, const long long* __restrict__ ei, int Ecnt)
{
  const int t = blockIdx.x * blockDim.x + threadIdx.x;
  if (t >= Ecnt * HEADS) return;
  const int h = t & 3;
  const int e = t >> 2;
  const int dst = (int)ei[Ecnt + e];
  const float pv = __expf(logits[t] - mx[dst * HEADS + h]);
  logits[t] = pv;
  unsafeAtomicAdd(&denom[dst * HEADS + h], pv);  // global_atomic_add_f32
}

// accum[dst] += xl[src] * (p / (denom[dst] + 1e-16))
__global__ __launch_bounds__(256) void aggregate_kernel(
    const float* __restrict__ xl, const float* __restrict__ p,
    const float* __restrict__ denom, const long long* __restrict__ ei,
    int Ecnt, float* __restrict__ accum)
{
  const int t = blockIdx.x * blockDim.x + threadIdx.x;
  if (t >= Ecnt * HEADS) return;
  const int h = t & 3;
  const int e = t >> 2;
  const int src = (int)ei[e];
  const int dst = (int)ei[Ecnt + e];
  const float alpha = p[t] / (denom[dst * HEADS + h] + 1e-16f);
  const float4* xs = reinterpret_cast<const float4*>(xl + src * HC + h * HID);
  float*        ad = accum + dst * HC + h * HID;
  #pragma unroll 4
  for (int c4 = 0; c4 < HID / 4; ++c4) {
    const float4 v = xs[c4];                       // b128 load, L2-resident
    unsafeAtomicAdd(&ad[c4 * 4 + 0], v.x * alpha);
    unsafeAtomicAdd(&ad[c4 * 4 + 1], v.y * alpha);
    unsafeAtomicAdd(&ad[c4 * 4 + 2], v.z * alpha);
    unsafeAtomicAdd(&ad[c4 * 4 + 3], v.w * alpha);
  }
}

// h_out = mean_heads(accum) + bias; optional gated residual with sigmoid(gate).
__global__ __launch_bounds__(256) void finalize_kernel(
    const float* __restrict__ accum, const float* __restrict__ bias,
    float* __restrict__ hbuf, int Nn, const float* __restrict__ gates, int gidx)
{
  const int i = blockIdx.x * blockDim.x + threadIdx.x;
  if (i >= Nn * HID) return;
  const int nid = i >> 6;
  const int c   = i & 63;
  const float* p = accum + nid * HC;
  float s = 0.25f * (p[c] + p[HID + c] + p[2 * HID + c] + p[3 * HID + c]) + bias[c];
  if (gates != nullptr) {
    const float g = 1.f / (1.f + __expf(-gates[gidx]));
    s = g * s + (1.f - g) * hbuf[i];   // residual uses pre-activation h
  }
  hbuf[i] = s;
}

static inline unsigned cdiv(long long a, long long b) { return (unsigned)((a + b - 1) / b); }

extern "C" void kernel_launch(void* const* d_in, const int* in_sizes, int n_in,
                              void* d_out, int out_size, void* d_ws, size_t ws_size,
                              hipStream_t stream)
{
  const float* x     = (const float*)d_in[0];
  const float* eattr = (const float*)d_in[1];
  const float* iWl   = (const float*)d_in[2];
  const float* ibl   = (const float*)d_in[3];
  const float* iWr   = (const float*)d_in[4];
  const float* ibr   = (const float*)d_in[5];
  const float* iWe   = (const float*)d_in[6];
  const float* iatt  = (const float*)d_in[7];
  const float* ibias = (const float*)d_in[8];
  const float* sWl   = (const float*)d_in[9];
  const float* sbl   = (const float*)d_in[10];
  const float* sWr   = (const float*)d_in[11];
  const float* sbr   = (const float*)d_in[12];
  const float* sWe   = (const float*)d_in[13];
  const float* satt  = (const float*)d_in[14];
  const float* sbias = (const float*)d_in[15];
  const float* gates = (const float*)d_in[16];
  const long long* ei = (const long long*)d_in[17];   // int64 edge_index (2,E)

  const int Nn   = in_sizes[0] / 128;   // 50000
  const int Ecnt = in_sizes[17] / 2;    // 800000

  float* out = (float*)d_out;           // doubles as the persistent h buffer

  // Workspace partition (floats): xl, xr, accum (N*256 each), logits (E*4),
  // mx (N*4), denom (N*4).  Total ~168 MB.
  float* xl     = (float*)d_ws;
  float* xr     = xl + (size_t)Nn * HC;
  float* accum  = xr + (size_t)Nn * HC;
  float* logits = accum + (size_t)Nn * HC;
  float* mx     = logits + (size_t)Ecnt * HEADS;
  float* denom  = mx + (size_t)Nn * HEADS;

  const dim3 gemmGrid(cdiv(Nn, 16), 2);
  const unsigned edgeBlocks = cdiv((long long)Ecnt * HEADS, 256);
  const unsigned nodeHCBlk  = cdiv((long long)Nn * HC, 256);
  const unsigned nodeHBlk   = cdiv((long long)Nn * HEADS, 256);
  const unsigned nodeCBlk   = cdiv((long long)Nn * HID, 256);

  for (int layer = 0; layer < 3; ++layer) {
    const int li = layer - 1;  // index into stacked params for layers 1,2
    const float* Wl_ = (layer == 0) ? iWl  : sWl  + (size_t)li * HID * HC;
    const float* bl_ = (layer == 0) ? ibl  : sbl  + (size_t)li * HC;
    const float* Wr_ = (layer == 0) ? iWr  : sWr  + (size_t)li * HID * HC;
    const float* br_ = (layer == 0) ? ibr  : sbr  + (size_t)li * HC;
    const float* We_ = (layer == 0) ? iWe  : sWe  + (size_t)li * 9 * HC;
    const float* at_ = (layer == 0) ? iatt : satt + (size_t)li * HEADS * HID;
    const float* bs_ = (layer == 0) ? ibias: sbias+ (size_t)li * HID;

    if (layer == 0) {
      gemm_wmma_kernel<128, false><<<gemmGrid, 256, 0, stream>>>(x, Wl_, bl_, xl, Nn);
      gemm_wmma_kernel<128, false><<<gemmGrid, 256, 0, stream>>>(x, Wr_, br_, xr, Nn);
    } else {
      gemm_wmma_kernel<64, true><<<gemmGrid, 256, 0, stream>>>(out, Wl_, bl_, xl, Nn);
      gemm_wmma_kernel<64, true><<<gemmGrid, 256, 0, stream>>>(out, Wr_, br_, xr, Nn);
    }

    init_kernel<<<nodeHCBlk, 256, 0, stream>>>(accum, mx, denom, Nn);
    edge_logits_kernel<<<edgeBlocks, 256, 0, stream>>>(
        xl, xr, eattr, We_, at_, ei, Ecnt, logits, mx);
    fix_mx_kernel<<<nodeHBlk, 256, 0, stream>>>(mx, Nn);
    edge_exp_kernel<<<edgeBlocks, 256, 0, stream>>>(logits, mx, denom, ei, Ecnt);
    aggregate_kernel<<<edgeBlocks, 256, 0, stream>>>(xl, logits, denom, ei, Ecnt, accum);
    finalize_kernel<<<nodeCBlk, 256, 0, stream>>>(
        accum, bs_, out, Nn, (layer == 0) ? nullptr : gates, li);
  }
}